// Contraction_627065225897
// MI455X (gfx1250) — compile-verified
//
#include <hip/hip_runtime.h>

// MACE-style symmetric contraction, fully fused for MI455X (gfx1250, wave32).
// B=128 atoms, F=128 channels, L=16, E=10, P3=23, P2=4, P1=1.
//
// Per block (256 thr = 8 waves): one atom; each wave owns a 16-channel tile.
// U3[p] slabs (16 KB) are double-buffered in LDS via async global->LDS DMA
// (GLOBAL_LOAD_ASYNC_TO_LDS_B128, ASYNCcnt), overlapped with 64 chained
// V_WMMA_F32_16X16X4_F32 per slab.

typedef float v2f __attribute__((ext_vector_type(2)));
typedef float v8f __attribute__((ext_vector_type(8)));

// Builtin's expected pointee type (from hipcc diagnostic): int vector_size(16)
typedef int v4i_t __attribute__((vector_size(16)));
typedef __attribute__((address_space(1))) void  gvoid_t;   // global
typedef __attribute__((address_space(3))) void  lvoid_t;   // LDS
typedef __attribute__((address_space(1))) v4i_t gv4i_t;
typedef __attribute__((address_space(3))) v4i_t lv4i_t;

#define B_ATOMS 128
#define F_CH    128
#define L_DIM   16
#define P3C     23
#define P2C     4

#if defined(__has_builtin)
#if __has_builtin(__builtin_amdgcn_global_load_async_to_lds_b128)
#define HAVE_ASYNC_LDS 1
#endif
#endif

#ifndef HAVE_ASYNC_LDS
#define HAVE_ASYNC_LDS 0
#endif

static __device__ inline v8f zero8() {
    v8f z;
#pragma unroll
    for (int i = 0; i < 8; ++i) z[i] = 0.0f;
    return z;
}

static __device__ inline void wait_async0() {
#if HAVE_ASYNC_LDS
#if __has_builtin(__builtin_amdgcn_s_wait_asynccnt)
    __builtin_amdgcn_s_wait_asynccnt(0);
#else
    asm volatile("s_wait_asynccnt 0x0" ::: "memory");
#endif
#endif
}

#if HAVE_ASYNC_LDS
static __device__ inline void async_copy_b128(const float* src, float* lds_dst) {
    __builtin_amdgcn_global_load_async_to_lds_b128(
        (gv4i_t*)(gvoid_t*)(void*)src,
        (lv4i_t*)(lvoid_t*)(void*)lds_dst,
        0, 0);
}
#endif

__global__ __launch_bounds__(256) void mace_contract_kernel(
    const float* __restrict__ x,              // [B,F,16]
    const long long* __restrict__ sorted_idx, // [B] (int64)
    const float* __restrict__ wmax,           // [E,P3,F]
    const float* __restrict__ w2,             // [E,P2,F]
    const float* __restrict__ w1,             // [E,1,F]
    const float* __restrict__ U3,             // [P3, 4096]
    const float* __restrict__ U2,             // [P2, 256]
    const float* __restrict__ U1,             // [16]
    float* __restrict__ out)                  // [B,F]
{
    __shared__ float ldsU[2][4096];           // double-buffered U3 slabs

    const int b    = blockIdx.x;              // atom
    const int wav  = threadIdx.x >> 5;        // 8 waves -> 8 f-tiles
    const int lane = threadIdx.x & 31;
    const int h    = lane >> 4;               // lane half
    const int n16  = lane & 15;
    const int f0   = wav * 16;

    const int e = (int)sorted_idx[b];

#if HAVE_ASYNC_LDS
    // Stage U3[p] slab into ldsU[buf]: 4096 floats, 256 threads, 4x b128 each.
    auto stage_async = [&](int p, int buf) {
#pragma unroll
        for (int i = 0; i < 4; ++i) {
            const int off = (threadIdx.x + i * 256) * 4;   // float idx, 16B aligned
            async_copy_b128(U3 + p * 4096 + off, &ldsU[buf][off]);
        }
    };
#endif

    // ---- A operand: x tile, rows = channels f0..f0+15, K = 0..15 (4 k-blocks)
    // f32 16x4 A layout: lanes 0-15 rows, VGPR j holds K = kb*4 + 2*h + j.
    const float* xb = x + (b * F_CH + f0) * L_DIM;
    v2f a[4];
#pragma unroll
    for (int kb = 0; kb < 4; ++kb)
        a[kb] = *(const v2f*)(xb + n16 * L_DIM + kb * 4 + 2 * h);

    // xv[v][lane] = x[b, f0+v+8h, n16]  (matches C/D row layout for row dots)
    float xv[8];
#pragma unroll
    for (int v = 0; v < 8; ++v)
        xv[v] = xb[(v + 8 * h) * L_DIM + n16];

    // =========== correlation 3: acc3[mt][f, l] = sum_p W3 * (x @ U3[p]^T) ====
    v8f acc3[16];
#pragma unroll
    for (int mt = 0; mt < 16; ++mt) acc3[mt] = zero8();

#if HAVE_ASYNC_LDS
    stage_async(0, 0);                        // prime the pipeline
#endif

#pragma unroll 1
    for (int p = 0; p < P3C; ++p) {
        const int buf = p & 1;

#if HAVE_ASYNC_LDS
        wait_async0();                        // own slab-p DMA landed in LDS
        __syncthreads();                      // everyone's landed; buf^1 free
        if (p + 1 < P3C)
            stage_async(p + 1, buf ^ 1);      // overlap next DMA with compute
#else
        __syncthreads();
        {
            const float4* src = (const float4*)(U3 + p * 4096);
            float4*       dst = (float4*)&ldsU[buf][0];
#pragma unroll
            for (int i = 0; i < 4; ++i)
                dst[threadIdx.x + i * 256] = src[threadIdx.x + i * 256];
        }
        __syncthreads();
#endif
        if (p + 2 < P3C)                      // speculative L2 warm (prefetch_b8)
            __builtin_prefetch(U3 + (p + 2) * 4096 + threadIdx.x * 16, 0, 1);

        // per-row scales: ws[v] = wmax[e, p, f0 + v + 8h]
        const float* wp = wmax + (e * P3C + p) * F_CH + f0 + 8 * h;
        float ws[8];
#pragma unroll
        for (int v = 0; v < 8; ++v) ws[v] = wp[v];

        const float* Up = &ldsU[buf][0];
#pragma unroll
        for (int mt = 0; mt < 16; ++mt) {
            v8f y = zero8();
#pragma unroll
            for (int kb = 0; kb < 4; ++kb) {
                // B[k, n] = U3[p, (mt*16 + n)*16 + k]; n = n16, k = kb*4+2h+j
                v2f bop = *(const v2f*)(Up + (mt * 16 + n16) * L_DIM +
                                        kb * 4 + 2 * h);
                y = __builtin_amdgcn_wmma_f32_16x16x4_f32(
                        false, a[kb], false, bop, (short)0, y, false, false);
            }
#pragma unroll
            for (int v = 0; v < 8; ++v)
                acc3[mt][v] = fmaf(ws[v], y[v], acc3[mt][v]);
        }
    }

    // =========== correlation 2 ============================================
    // out2[f, m] = sum_l acc3[m][f,l]*x[f,l]  +  sum_p W2[e,p,f]*(x @ U2[p]^T)
    v8f out2 = zero8();

    // row-wise dot of each m-tile with x, reduced across the 16-lane half
#pragma unroll
    for (int mt = 0; mt < 16; ++mt) {
#pragma unroll
        for (int v = 0; v < 8; ++v) {
            float t = acc3[mt][v] * xv[v];
            t += __shfl_xor(t, 1);
            t += __shfl_xor(t, 2);
            t += __shfl_xor(t, 4);
            t += __shfl_xor(t, 8);
            if (n16 == mt) out2[v] += t;   // place at column m = mt
        }
    }

    // U2 part via 16 more WMMAs (B operand straight from global; U2 is 4 KB)
#pragma unroll
    for (int p = 0; p < P2C; ++p) {
        v8f z = zero8();
#pragma unroll
        for (int kb = 0; kb < 4; ++kb) {
            v2f bop = *(const v2f*)(U2 + p * 256 + n16 * L_DIM + kb * 4 + 2 * h);
            z = __builtin_amdgcn_wmma_f32_16x16x4_f32(
                    false, a[kb], false, bop, (short)0, z, false, false);
        }
        const float* wp2 = w2 + (e * P2C + p) * F_CH + f0 + 8 * h;
#pragma unroll
        for (int v = 0; v < 8; ++v)
            out2[v] = fmaf(wp2[v], z[v], out2[v]);
    }

    // =========== correlation 1 + final row dot ============================
    // out[b,f] = sum_l (out2[f,l] + w1[e,f]*U1[l]) * x[f,l]
    const float u1v = U1[n16];
    const float* wp1 = w1 + e * F_CH + f0 + 8 * h;
#pragma unroll
    for (int v = 0; v < 8; ++v) {
        float t = (out2[v] + wp1[v] * u1v) * xv[v];
        t += __shfl_xor(t, 1);
        t += __shfl_xor(t, 2);
        t += __shfl_xor(t, 4);
        t += __shfl_xor(t, 8);
        if (n16 == 0)
            out[b * F_CH + f0 + v + 8 * h] = t;
    }
}

extern "C" void kernel_launch(void* const* d_in, const int* in_sizes, int n_in,
                              void* d_out, int out_size, void* d_ws, size_t ws_size,
                              hipStream_t stream) {
    // setup_inputs order: x, bincount, sorted_indices, weights_max, w2, w1, U3, U2, U1
    const float*     x    = (const float*)d_in[0];
    const long long* sidx = (const long long*)d_in[2];   // int64 indices
    const float*     wmax = (const float*)d_in[3];
    const float*     w2   = (const float*)d_in[4];
    const float*     w1   = (const float*)d_in[5];
    const float*     U3   = (const float*)d_in[6];
    const float*     U2   = (const float*)d_in[7];
    const float*     U1   = (const float*)d_in[8];
    float*           out  = (float*)d_out;

    hipLaunchKernelGGL(mace_contract_kernel, dim3(B_ATOMS), dim3(256), 0, stream,
                       x, sidx, wmax, w2, w1, U3, U2, U1, out);
}